// Seq2Seq_32384053412319
// MI455X (gfx1250) — compile-verified
//
#include <hip/hip_runtime.h>
#include <hip/hip_bf16.h>
#include <math.h>

// ---------------------------------------------------------------------------
// Seq2Seq GRU encoder/decoder with attention for MI455X (gfx1250, wave32).
//  - Batchable input-gate GEMMs run on V_WMMA_F32_16X16X4_F32.
//  - Sequential recurrence runs as a persistent 32-block kernel with a
//    device-wide sense-reversal barrier; every mat-vec is split across
//    4096 threads with float4 (b128) loads and wave shuffles.
// ---------------------------------------------------------------------------

typedef __attribute__((ext_vector_type(2))) float v2f;
typedef __attribute__((ext_vector_type(8))) float v8f;

#define Hh   1024
#define H3   3072
#define LL   50
#define TT   128
#define NPAD 64

// workspace offsets (in floats)
#define OFF_XEMB   0u        // 64 x 1024 gathered word embeddings (padded)
#define OFF_DEMB   65536u    // 64 x 1024 gathered decoder embeddings (padded)
#define OFF_GIENC  131072u   // 3072 x 64  enc_Wih @ X^T + bih
#define OFF_CE     327680u   // 1024 x 64  comb_W[:, :H] @ E^T
#define OFF_ENC    393216u   // 50 x 1024  encoder outputs
#define OFF_HA     444416u   // h double buffer A
#define OFF_HB     445440u   // h double buffer B
#define OFF_AW     446464u   // 64 attention logits
#define OFF_CTX    446528u   // 1024 context
#define OFF_O      447552u   // 1024 combined input o
#define OFF_LOGI   448576u   // 128 logits
#define OFF_SYNC   448704u   // 8 uints: global barrier state

#define GBLK 32
#define GTHR 128

__device__ __forceinline__ float dot4(float4 a, float4 b) {
    return a.x * b.x + a.y * b.y + a.z * b.z + a.w * b.w;
}
__device__ __forceinline__ float sigf(float x) { return 1.f / (1.f + expf(-x)); }

// ------------------------- device-wide barrier -----------------------------
__device__ __forceinline__ void gbar(unsigned* cnt, volatile unsigned* gen) {
    __syncthreads();
    if (threadIdx.x == 0) {
        __threadfence();
        unsigned g = *gen;
        if (atomicAdd(cnt, 1u) == (unsigned)(GBLK - 1)) {
            *(volatile unsigned*)cnt = 0u;
            __threadfence();
            atomicAdd((unsigned*)gen, 1u);
        } else {
            while (*gen == g) { __builtin_amdgcn_s_sleep(1); }
        }
        __threadfence();
    }
    __syncthreads();
}

// block-0 reductions over 128 threads
__device__ __forceinline__ float blk_max(float v, float* red, int tid) {
    red[tid] = v; __syncthreads();
    for (int s = 64; s > 0; s >>= 1) {
        if (tid < s) red[tid] = fmaxf(red[tid], red[tid + s]);
        __syncthreads();
    }
    float r = red[0]; __syncthreads(); return r;
}
__device__ __forceinline__ float blk_sum(float v, float* red, int tid) {
    red[tid] = v; __syncthreads();
    for (int s = 64; s > 0; s >>= 1) {
        if (tid < s) red[tid] = red[tid] + red[tid + s];
        __syncthreads();
    }
    float r = red[0]; __syncthreads(); return r;
}

// --------------------------- helper kernels --------------------------------
__global__ __launch_bounds__(32) void init_sync(unsigned* sync) {
    if (threadIdx.x < 8) sync[threadIdx.x] = 0u;
}

// mode 0: row r uses ids[r]; mode 1 (decoder shift): row 0 -> token 1 (O_TOK),
// row r -> ids[r-1]. Rows >= 50 are zero-padded.
__global__ __launch_bounds__(256) void gather_rows(const int* __restrict__ ids,
                                                   const float* __restrict__ tab,
                                                   float* __restrict__ dst, int mode) {
    int r = blockIdx.x;
    int tok = -1;
    if (r < LL) tok = (mode == 0) ? ids[r] : ((r == 0) ? 1 : ids[r - 1]);
    const float* src = (tok >= 0) ? (tab + (size_t)tok * Hh) : nullptr;
    for (int c = threadIdx.x; c < Hh; c += blockDim.x)
        dst[(size_t)r * Hh + c] = (tok >= 0) ? src[c] : 0.f;
}

// --------------------------- WMMA GEMM (N = 64) ----------------------------
// out(M x 64) = W(M x K, row stride ldw) @ X(64 x K, row stride 1024)^T [+ bias]
// One wave per 16-row stripe; the A fragment is reused across 4 N-tiles.
// f32 WMMA operand layout (ISA 7.12.2):
//   A 16x4 : lane%16 = M, reg j holds K = j + 2*(lane>=16)
//   B 4x16 : lane%16 = N, reg j holds K = j + 2*(lane>=16)
//   C 16x16: reg r, lanes 0-15 -> (M=r, N=lane), lanes 16-31 -> (M=r+8, N=lane-16)
__global__ __launch_bounds__(128) void wmma_gemm_n64(const float* __restrict__ W, int ldw,
                                                     const float* __restrict__ X,
                                                     const float* __restrict__ bias,
                                                     float* __restrict__ out, int K) {
    const int wave = (blockIdx.x * blockDim.x + threadIdx.x) >> 5;
    const int lane = threadIdx.x & 31;
    const int hi   = lane >> 4;
    const int l16  = lane & 15;
    const int m    = wave * 16 + l16;

    v8f acc0 = {}, acc1 = {}, acc2 = {}, acc3 = {};
    const float* wrow = W + (size_t)m * ldw;
    const float* x0 = X + (size_t)(0 * 16 + l16) * 1024;
    const float* x1 = X + (size_t)(1 * 16 + l16) * 1024;
    const float* x2 = X + (size_t)(2 * 16 + l16) * 1024;
    const float* x3 = X + (size_t)(3 * 16 + l16) * 1024;

    for (int k = 0; k < K; k += 4) {
        const int ka = k + (hi << 1);
        v2f a;  a.x  = wrow[ka]; a.y  = wrow[ka + 1];
        v2f b0; b0.x = x0[ka];   b0.y = x0[ka + 1];
        v2f b1; b1.x = x1[ka];   b1.y = x1[ka + 1];
        v2f b2; b2.x = x2[ka];   b2.y = x2[ka + 1];
        v2f b3; b3.x = x3[ka];   b3.y = x3[ka + 1];
        acc0 = __builtin_amdgcn_wmma_f32_16x16x4_f32(false, a, false, b0, (short)0, acc0, false, false);
        acc1 = __builtin_amdgcn_wmma_f32_16x16x4_f32(false, a, false, b1, (short)0, acc1, false, false);
        acc2 = __builtin_amdgcn_wmma_f32_16x16x4_f32(false, a, false, b2, (short)0, acc2, false, false);
        acc3 = __builtin_amdgcn_wmma_f32_16x16x4_f32(false, a, false, b3, (short)0, acc3, false, false);
    }

#pragma unroll
    for (int r = 0; r < 8; ++r) {
        const int mm = wave * 16 + r + (hi << 3);
        const float bv = bias ? bias[mm] : 0.f;
        const size_t o = (size_t)mm * 64 + l16;
        out[o]      = acc0[r] + bv;
        out[o + 16] = acc1[r] + bv;
        out[o + 32] = acc2[r] + bv;
        out[o + 48] = acc3[r] + bv;
    }
}

// ----------------------- persistent sequential kernel ----------------------
__global__ __launch_bounds__(GTHR) void seq_kernel(
    float* __restrict__ ws,
    const int* __restrict__ tag_ids,
    const float* __restrict__ enc_Whh, const float* __restrict__ enc_bhh,
    const float* __restrict__ attn_W,  const float* __restrict__ attn_b,
    const float* __restrict__ comb_W,  const float* __restrict__ comb_b,
    const float* __restrict__ dec_Wih, const float* __restrict__ dec_Whh,
    const float* __restrict__ dec_bih, const float* __restrict__ dec_bhh,
    const float* __restrict__ out_W,   const float* __restrict__ out_b,
    float* __restrict__ out) {

    const int tid = threadIdx.x;
    const int gid = blockIdx.x * GTHR + tid;
    const int l4  = gid & 3;
    const int i4  = gid >> 2;           // 0..1023

    unsigned* cnt          = (unsigned*)(ws + OFF_SYNC);
    volatile unsigned* gen = (volatile unsigned*)(ws + OFF_SYNC) + 1;

    float* GiEnc  = ws + OFF_GIENC;
    float* Ce     = ws + OFF_CE;
    float* encO   = ws + OFF_ENC;
    float* Demb   = ws + OFF_DEMB;
    float* aw_ws  = ws + OFF_AW;
    float* ctx_ws = ws + OFF_CTX;
    float* o_ws   = ws + OFF_O;
    float* lg_ws  = ws + OFF_LOGI;
    float* hc = ws + OFF_HA;
    float* hnx = ws + OFF_HB;

    __shared__ float s_ew[64];
    __shared__ float s_mx, s_inv;
    __shared__ float s_red[TT];
    __shared__ float s_lp[TT];

    float loss = 0.f;

    if (gid < Hh) hc[gid] = 0.f;
    gbar(cnt, gen);

    // ------------------------------ encoder -------------------------------
    for (int t = 0; t < LL; ++t) {
        const float4* w0 = (const float4*)(enc_Whh + (size_t)i4 * Hh) + l4 * 64;
        const float4* w1 = (const float4*)(enc_Whh + (size_t)(i4 + Hh) * Hh) + l4 * 64;
        const float4* w2 = (const float4*)(enc_Whh + (size_t)(i4 + 2 * Hh) * Hh) + l4 * 64;
        const float4* hv = (const float4*)hc + l4 * 64;
        float s0 = 0.f, s1 = 0.f, s2 = 0.f;
#pragma unroll 4
        for (int kk = 0; kk < 64; ++kk) {
            float4 h4 = hv[kk];
            s0 += dot4(w0[kk], h4); s1 += dot4(w1[kk], h4); s2 += dot4(w2[kk], h4);
        }
        s0 += __shfl_down(s0, 2, 4); s0 += __shfl_down(s0, 1, 4);
        s1 += __shfl_down(s1, 2, 4); s1 += __shfl_down(s1, 1, 4);
        s2 += __shfl_down(s2, 2, 4); s2 += __shfl_down(s2, 1, 4);
        if (l4 == 0) {
            float hprev = hc[i4];
            float gir = GiEnc[(size_t)i4 * 64 + t];
            float giz = GiEnc[(size_t)(i4 + Hh) * 64 + t];
            float gin = GiEnc[(size_t)(i4 + 2 * Hh) * 64 + t];
            float ghr = s0 + enc_bhh[i4];
            float ghz = s1 + enc_bhh[i4 + Hh];
            float ghn = s2 + enc_bhh[i4 + 2 * Hh];
            float r = sigf(gir + ghr), z = sigf(giz + ghz);
            float n = tanhf(gin + r * ghn);
            float h2 = (1.f - z) * n + z * hprev;
            hnx[i4] = h2;
            encO[(size_t)t * Hh + i4] = h2;
        }
        gbar(cnt, gen);
        { float* tmp = hc; hc = hnx; hnx = tmp; }
    }

    // ------------------------------ decoder -------------------------------
    for (int t = 0; t < LL; ++t) {
        const float* e = Demb + (size_t)t * Hh;

        // A: attention logits a[j] = attn_W[j] . [e ; h] + attn_b[j]
        if (gid < 800) {
            int j = gid >> 4, l = gid & 15;
            const float4* wr = (const float4*)(attn_W + (size_t)j * 2048 + l * 128);
            const float4* xv = (l < 8) ? ((const float4*)e + l * 32)
                                       : ((const float4*)hc + (l - 8) * 32);
            float s = 0.f;
#pragma unroll 4
            for (int kk = 0; kk < 32; ++kk) s += dot4(wr[kk], xv[kk]);
            s += __shfl_down(s, 8, 16); s += __shfl_down(s, 4, 16);
            s += __shfl_down(s, 2, 16); s += __shfl_down(s, 1, 16);
            if (l == 0) aw_ws[j] = s + attn_b[j];
        }
        gbar(cnt, gen);

        // B: softmax (redundant per block) + ctx = aw @ enc_outs
        if (tid == 0) {
            float m = -1e30f;
            for (int j = 0; j < LL; ++j) m = fmaxf(m, aw_ws[j]);
            s_mx = m;
        }
        __syncthreads();
        if (tid < 64) s_ew[tid] = (tid < LL) ? expf(aw_ws[tid] - s_mx) : 0.f;
        __syncthreads();
        if (tid == 0) {
            float s = 0.f;
            for (int j = 0; j < LL; ++j) s += s_ew[j];
            s_inv = 1.f / s;
        }
        __syncthreads();
        if (gid < Hh) {
            float s = 0.f;
            for (int j = 0; j < LL; ++j) s += s_ew[j] * encO[(size_t)j * Hh + gid];
            ctx_ws[gid] = s * s_inv;
        }
        gbar(cnt, gen);

        // C: o = relu(Ce[:,t] + comb_W[:,H:] @ ctx + comb_b)
        {
            const float4* wr = (const float4*)(comb_W + (size_t)i4 * 2048 + Hh) + l4 * 64;
            const float4* xv = (const float4*)ctx_ws + l4 * 64;
            float s = 0.f;
#pragma unroll 4
            for (int kk = 0; kk < 64; ++kk) s += dot4(wr[kk], xv[kk]);
            s += __shfl_down(s, 2, 4); s += __shfl_down(s, 1, 4);
            if (l4 == 0) {
                float v = s + Ce[(size_t)i4 * 64 + t] + comb_b[i4];
                o_ws[i4] = fmaxf(v, 0.f);
            }
        }
        gbar(cnt, gen);

        // D: GRU cell (gi = dec_Wih@o + bih, gh = dec_Whh@h + bhh)
        {
            const float4* a0 = (const float4*)(dec_Wih + (size_t)i4 * Hh) + l4 * 64;
            const float4* a1 = (const float4*)(dec_Wih + (size_t)(i4 + Hh) * Hh) + l4 * 64;
            const float4* a2 = (const float4*)(dec_Wih + (size_t)(i4 + 2 * Hh) * Hh) + l4 * 64;
            const float4* b0 = (const float4*)(dec_Whh + (size_t)i4 * Hh) + l4 * 64;
            const float4* b1 = (const float4*)(dec_Whh + (size_t)(i4 + Hh) * Hh) + l4 * 64;
            const float4* b2 = (const float4*)(dec_Whh + (size_t)(i4 + 2 * Hh) * Hh) + l4 * 64;
            const float4* ov = (const float4*)o_ws + l4 * 64;
            const float4* hv = (const float4*)hc + l4 * 64;
            float gi0 = 0.f, gi1 = 0.f, gi2 = 0.f, gh0 = 0.f, gh1 = 0.f, gh2 = 0.f;
#pragma unroll 2
            for (int kk = 0; kk < 64; ++kk) {
                float4 o4 = ov[kk], h4 = hv[kk];
                gi0 += dot4(a0[kk], o4); gi1 += dot4(a1[kk], o4); gi2 += dot4(a2[kk], o4);
                gh0 += dot4(b0[kk], h4); gh1 += dot4(b1[kk], h4); gh2 += dot4(b2[kk], h4);
            }
            gi0 += __shfl_down(gi0, 2, 4); gi0 += __shfl_down(gi0, 1, 4);
            gi1 += __shfl_down(gi1, 2, 4); gi1 += __shfl_down(gi1, 1, 4);
            gi2 += __shfl_down(gi2, 2, 4); gi2 += __shfl_down(gi2, 1, 4);
            gh0 += __shfl_down(gh0, 2, 4); gh0 += __shfl_down(gh0, 1, 4);
            gh1 += __shfl_down(gh1, 2, 4); gh1 += __shfl_down(gh1, 1, 4);
            gh2 += __shfl_down(gh2, 2, 4); gh2 += __shfl_down(gh2, 1, 4);
            if (l4 == 0) {
                float ir = gi0 + dec_bih[i4];
                float iz = gi1 + dec_bih[i4 + Hh];
                float in_ = gi2 + dec_bih[i4 + 2 * Hh];
                float hr = gh0 + dec_bhh[i4];
                float hz = gh1 + dec_bhh[i4 + Hh];
                float hn_ = gh2 + dec_bhh[i4 + 2 * Hh];
                float r = sigf(ir + hr), z = sigf(iz + hz);
                float n = tanhf(in_ + r * hn_);
                hnx[i4] = (1.f - z) * n + z * hc[i4];
            }
        }
        gbar(cnt, gen);

        // E: logits = out_W @ h2 + out_b
        {
            int row = gid >> 5, l = gid & 31;
            const float4* wr = (const float4*)(out_W + (size_t)row * Hh) + l * 8;
            const float4* hv = (const float4*)hnx + l * 8;
            float s = 0.f;
#pragma unroll
            for (int kk = 0; kk < 8; ++kk) s += dot4(wr[kk], hv[kk]);
            s += __shfl_down(s, 16); s += __shfl_down(s, 8); s += __shfl_down(s, 4);
            s += __shfl_down(s, 2);  s += __shfl_down(s, 1);
            if (l == 0) lg_ws[row] = s + out_b[row];
        }
        gbar(cnt, gen);

        // F: block 0 computes logp (double log_softmax) + NLL, writes output
        if (blockIdx.x == 0) {
            float x  = lg_ws[tid];
            float m1 = blk_max(x, s_red, tid);
            float z1 = blk_sum(expf(x - m1), s_red, tid);
            float lp = x - m1 - logf(z1);
            s_lp[tid] = lp;
            float m2 = blk_max(lp, s_red, tid);
            float z2 = blk_sum(expf(lp - m2), s_red, tid);
            float lse2 = m2 + logf(z2);
            __syncthreads();
            out[(size_t)t * TT + tid] = lp;
            if (tid == 0) {
                int tg = tag_ids[t];
                if (tg != 0) loss += lse2 - s_lp[tg];
            }
        }
        { float* tmp = hc; hc = hnx; hnx = tmp; }
    }

    if (blockIdx.x == 0 && tid == 0) out[(size_t)LL * TT] = loss;
}

// ------------------------------- launcher ----------------------------------
extern "C" void kernel_launch(void* const* d_in, const int* in_sizes, int n_in,
                              void* d_out, int out_size, void* d_ws, size_t ws_size,
                              hipStream_t stream) {
    (void)in_sizes; (void)n_in; (void)out_size; (void)ws_size;
    const int*   input_ids = (const int*)d_in[0];
    const int*   tag_ids   = (const int*)d_in[1];
    const float* word_embed = (const float*)d_in[2];
    const float* enc_Wih   = (const float*)d_in[3];
    const float* enc_Whh   = (const float*)d_in[4];
    const float* enc_bih   = (const float*)d_in[5];
    const float* enc_bhh   = (const float*)d_in[6];
    const float* dec_embed = (const float*)d_in[7];
    const float* attn_W    = (const float*)d_in[8];
    const float* attn_b    = (const float*)d_in[9];
    const float* comb_W    = (const float*)d_in[10];
    const float* comb_b    = (const float*)d_in[11];
    const float* dec_Wih   = (const float*)d_in[12];
    const float* dec_Whh   = (const float*)d_in[13];
    const float* dec_bih   = (const float*)d_in[14];
    const float* dec_bhh   = (const float*)d_in[15];
    const float* out_W     = (const float*)d_in[16];
    const float* out_b     = (const float*)d_in[17];
    float* ws  = (float*)d_ws;
    float* out = (float*)d_out;

    init_sync<<<dim3(1), dim3(32), 0, stream>>>((unsigned*)(ws + OFF_SYNC));
    gather_rows<<<dim3(64), dim3(256), 0, stream>>>(input_ids, word_embed, ws + OFF_XEMB, 0);
    gather_rows<<<dim3(64), dim3(256), 0, stream>>>(tag_ids, dec_embed, ws + OFF_DEMB, 1);
    // GiEnc = enc_Wih (3072x1024) @ Xemb^T + bih  -> 192 waves
    wmma_gemm_n64<<<dim3(48), dim3(128), 0, stream>>>(enc_Wih, 1024, ws + OFF_XEMB,
                                                      enc_bih, ws + OFF_GIENC, 1024);
    // Ce = comb_W[:, :1024] (1024x1024, ld 2048) @ Demb^T -> 64 waves
    wmma_gemm_n64<<<dim3(16), dim3(128), 0, stream>>>(comb_W, 2048, ws + OFF_DEMB,
                                                      (const float*)nullptr, ws + OFF_CE, 1024);
    seq_kernel<<<dim3(GBLK), dim3(GTHR), 0, stream>>>(
        ws, tag_ids, enc_Whh, enc_bhh, attn_W, attn_b, comb_W, comb_b,
        dec_Wih, dec_Whh, dec_bih, dec_bhh, out_W, out_b, out);
}